// Hadamard_71150428226261
// MI455X (gfx1250) — compile-verified
//
#include <hip/hip_runtime.h>

typedef __attribute__((ext_vector_type(2))) float v2f;
typedef __attribute__((ext_vector_type(8))) float v8f;

// Problem constants (from reference): D=3, L=8, index=2, batch=4096.
constexpr int DQ    = 3;
constexpr int LEFT  = 9;     // D^index
constexpr int RIGHT = 243;   // D^(L-index-1)
constexpr int BATCH = 4096;

// W = [[Mr, -Mi], [Mi, Mr]]  (real 6x6 expansion of complex 3x3 gate),
// zero-padded to 16 rows x 8 K-cols for the WMMA.
__device__ __forceinline__ float wval(const float* __restrict__ Mf, int n, int k) {
    if (n >= 6 || k >= 6) return 0.0f;
    const int i = (n >= 3) ? n - 3 : n;
    const int j = (k >= 3) ? k - 3 : k;
    const float re = Mf[2 * (i * 3 + j) + 0];
    const float im = Mf[2 * (i * 3 + j) + 1];
    const bool nre = (n < 3), kre = (k < 3);
    if (nre && kre)  return re;    // Mr
    if (nre && !kre) return -im;   // -Mi
    if (!nre && kre) return im;    // Mi
    return re;                     // Mr
}

__global__ __launch_bounds__(256) void qudit_gate_wmma(
    const float* __restrict__ x_re,
    const float* __restrict__ x_im,
    const float* __restrict__ Mf,
    float* __restrict__ out)
{
    const int lane = threadIdx.x & 31;
    const int wave = threadIdx.x >> 5;
    const int col  = lane & 15;   // batch column within tile / A row
    const int half = lane >> 4;   // K-pair selector per ISA A/B layout

    const int p = blockIdx.x;            // (a, r) pair index, 0..2186
    const int a = p / RIGHT;
    const int r = p - a * RIGHT;

    // element offsets of the three j-slices (same for re, im, out planes)
    const long rb0 = ((long)(a * DQ + 0) * RIGHT + r) * (long)BATCH;
    const long rb1 = ((long)(a * DQ + 1) * RIGHT + r) * (long)BATCH;
    const long rb2 = ((long)(a * DQ + 2) * RIGHT + r) * (long)BATCH;

    const float* __restrict__ sr0 = x_re + rb0;
    const float* __restrict__ sr1 = x_re + rb1;
    const float* __restrict__ sr2 = x_re + rb2;
    const float* __restrict__ si0 = x_im + rb0;
    const float* __restrict__ si1 = x_im + rb1;
    const float* __restrict__ si2 = x_im + rb2;

    // ---- A operand (constant per wave): 16x4 f32 layout:
    //   VGPR0: lanes0-15 -> K=kA, lanes16-31 -> K=kA+2 (kA base 0 / 2)
    //   row M = lane & 15
    const int kA = half ? 2 : 0;
    v2f A0, A1;
    A0.x = wval(Mf, col, kA + 0);
    A0.y = wval(Mf, col, kA + 1);
    A1.x = wval(Mf, col, kA + 4);   // K = 4..7 (6,7 are zero pad)
    A1.y = wval(Mf, col, kA + 5);

    // ---- B operand stream selection (K rows = components xr0..xr2, xi0..xi2)
    const float* __restrict__ pB00 = half ? sr2 : sr0;  // K=2 : K=0
    const float* __restrict__ pB01 = half ? si0 : sr1;  // K=3 : K=1
    // second WMMA rows: K=4 -> xi1, K=5 -> xi2, K=6,7 -> 0

    // batch range for this wave: 256 elements = 16 tiles of 16
    const int chunk  = blockIdx.y * 8 + wave;       // 0..15
    const int bbase0 = chunk * (BATCH / 16);

    float2* __restrict__ o0 = reinterpret_cast<float2*>(out) + rb0;
    float2* __restrict__ o1 = reinterpret_cast<float2*>(out) + rb1;
    float2* __restrict__ o2 = reinterpret_cast<float2*>(out) + rb2;

    for (int t = 0; t < 16; ++t) {
        const int b = bbase0 + t * 16 + col;

        v2f B0, B1;
        B0.x = pB00[b];
        B0.y = pB01[b];
        const float v0 = si1[b];          // both halves hit same 64B lines
        const float v1 = si2[b];
        B1.x = half ? 0.0f : v0;          // K=6 pad for upper half
        B1.y = half ? 0.0f : v1;          // K=7 pad for upper half

        v8f c = {};
        c = __builtin_amdgcn_wmma_f32_16x16x4_f32(
                false, A0, false, B0, (short)0, c, false, false);
        c = __builtin_amdgcn_wmma_f32_16x16x4_f32(
                false, A1, false, B1, (short)0, c, false, false);

        // D rows 0..5 (lanes 0-15) = [re0,re1,re2,im0,im1,im2] for batch b.
        // re_i / im_i are adjacent floats of interleaved complex64 -> float2.
        if (lane < 16) {
            o0[b] = make_float2(c[0], c[3]);
            o1[b] = make_float2(c[1], c[4]);
            o2[b] = make_float2(c[2], c[5]);
        }
    }
}

extern "C" void kernel_launch(void* const* d_in, const int* in_sizes, int n_in,
                              void* d_out, int out_size, void* d_ws, size_t ws_size,
                              hipStream_t stream) {
    const float* x_re = (const float*)d_in[0];
    const float* x_im = (const float*)d_in[1];
    const float* Mf   = (const float*)d_in[2];   // 3x3 complex64 = 18 floats
    float*       out  = (float*)d_out;           // N*B complex64 interleaved

    dim3 grid(LEFT * RIGHT, 2);                  // 2187 x 2 blocks
    qudit_gate_wmma<<<grid, 256, 0, stream>>>(x_re, x_im, Mf, out);
}